// NeuralPoints_1443109012011
// MI455X (gfx1250) — compile-verified
//
#include <hip/hip_runtime.h>
#include <hip/hip_bf16.h>

typedef __attribute__((ext_vector_type(2))) float v2f;
typedef __attribute__((ext_vector_type(4))) float v4f;
typedef __attribute__((ext_vector_type(8))) float v8f;

// ---------------------------------------------------------------------------
// Kernel 1: w2pers via V_WMMA_F32_16X16X4_F32.
// One wave transforms 16 points: D(16x16) = A(16x4: shift | 0) x B(4x16: R | 0).
// Columns 0..2 of D are camera-space xyz; perspective divide uses z pulled
// cross-lane from column 2 via wave32 shfl, then a single v_rcp_f32 + mul
// (instead of the IEEE div chain) and lanes n<3 store pers[p][n].
// ---------------------------------------------------------------------------
__global__ void np_pers_wmma_kernel(const float* __restrict__ xyz,
                                    const float* __restrict__ camrot,   // [3,3] row-major (B==1)
                                    const float* __restrict__ campos,   // [3]
                                    float* __restrict__ pers,           // [N,3]
                                    int npts) {
  const int lane = threadIdx.x & 31;
  const int wid  = (int)blockIdx.x * (blockDim.x >> 5) + (threadIdx.x >> 5);
  const int base = wid * 16;
  if (base >= npts) return;                 // wave-uniform branch: EXEC stays all-1s

  const int m    = lane & 15;               // A row / B column / D column index
  const int half = lane >> 4;

  const float cx = campos[0], cy = campos[1], cz = campos[2];

  int p = base + m;
  if (p >= npts) p = npts - 1;              // clamp (no-op for N % 16 == 0)
  const float sx = xyz[p * 3 + 0] - cx;
  const float sy = xyz[p * 3 + 1] - cy;
  const float sz = xyz[p * 3 + 2] - cz;

  // A 16x4 (MxK) f32 layout: lanes 0-15 hold K=0 (v0), K=1 (v1);
  //                          lanes 16-31 hold K=2 (v0), K=3 (v1).
  v2f A;
  A.x = half ? sz : sx;
  A.y = half ? 0.0f : sy;

  // B 4x16 (KxN): row k, col n = camrot[k*3+n] for k<3 && n<3, else 0.
  // Layout mirrors A: v0 = rows K=0 / K=2 per lane-half, v1 = rows K=1 / K=3.
  const int n = m;
  const bool nc = (n < 3);
  v2f Bm;
  Bm.x = nc ? (half ? camrot[2 * 3 + n] : camrot[0 * 3 + n]) : 0.0f;
  Bm.y = (nc && !half) ? camrot[1 * 3 + n] : 0.0f;

  v8f C = {};
  // (neg_a, A, neg_b, B, c_mod, C, reuse_a, reuse_b)
  v8f D = __builtin_amdgcn_wmma_f32_16x16x4_f32(false, A, false, Bm,
                                                (short)0, C, false, false);

  // D 16x16 f32 layout: VGPR v, lanes 0-15 -> M=v, N=lane; lanes 16-31 -> M=v+8.
#pragma unroll
  for (int v = 0; v < 8; ++v) {
    const float d = D[v];
    const float z = __shfl(d, (lane & 16) | 2, 32);   // D[m][2] for this lane's m
    const float r = __builtin_amdgcn_rcpf(z + 1e-9f); // single v_rcp_f32
    const float val = (n == 2) ? z : d * r;
    const int pm = base + half * 8 + v;
    if (nc && pm < npts)
      __builtin_nontemporal_store(val, pers + pm * 3 + n);
  }
}

// ---------------------------------------------------------------------------
// Kernel 2: embedding gather, lane-per-float4-chunk (F == 32 -> 8 lanes/slot).
// Reads: 8 lanes cover one contiguous 128B feats row (L2-resident table).
// Writes: consecutive slots across the wave -> fully coalesced 128B NT stores
// (streaming output must not evict the gather tables from L2).
// ---------------------------------------------------------------------------
__global__ void np_gather_emb_kernel(const float* __restrict__ feats,
                                     const int* __restrict__ pidx,
                                     float* __restrict__ out_emb,
                                     int slots) {
  const int gid  = (int)blockIdx.x * blockDim.x + threadIdx.x;
  const int slot = gid >> 3;                 // F==32: 8 float4 chunks per slot
  const int c    = (gid & 7) << 2;
  if (slot >= slots) return;

  int p = pidx[slot];
  p = p < 0 ? 0 : p;                         // reference clips (mask is separate)

  // Warm WGP$/L2 for a future slot's scattered row (global_prefetch_b8).
  if ((gid & 7) == 0) {
    const int ps = slot + 64;
    if (ps < slots) {
      int pp = pidx[ps];
      pp = pp < 0 ? 0 : pp;
      __builtin_prefetch(feats + (size_t)pp * 32, 0, 0);
    }
  }

  const v4f v = *reinterpret_cast<const v4f*>(feats + (size_t)p * 32 + c);
  __builtin_nontemporal_store(v, reinterpret_cast<v4f*>(out_emb + (size_t)slot * 32 + c));
}

// ---------------------------------------------------------------------------
// Kernel 3: small per-slot gathers: color(3), dir(3), conf(1), xyz(3),
// pers(3) + mask. One thread per slot; tables are L2-resident (72MB < 192MB).
// Outputs are streamed with NT stores.
// ---------------------------------------------------------------------------
__global__ void np_gather_misc_kernel(const int* __restrict__ pidx,
                                      const float* __restrict__ pcolor,
                                      const float* __restrict__ pdirv,
                                      const float* __restrict__ pconf,
                                      const float* __restrict__ xyz,
                                      const float* __restrict__ pers,
                                      float* __restrict__ o_color,
                                      float* __restrict__ o_dir,
                                      float* __restrict__ o_conf,
                                      float* __restrict__ o_xyz,
                                      float* __restrict__ o_pers,
                                      float* __restrict__ o_mask,
                                      int slots) {
  const int s = (int)blockIdx.x * blockDim.x + threadIdx.x;
  if (s >= slots) return;

  const int pr = pidx[s];
  const float mk = (pr >= 0) ? 1.0f : 0.0f;
  const size_t p  = (size_t)(pr < 0 ? 0 : pr);
  const size_t p3 = p * 3;
  const size_t s3 = (size_t)s * 3;

  __builtin_nontemporal_store(pcolor[p3 + 0], o_color + s3 + 0);
  __builtin_nontemporal_store(pcolor[p3 + 1], o_color + s3 + 1);
  __builtin_nontemporal_store(pcolor[p3 + 2], o_color + s3 + 2);

  __builtin_nontemporal_store(pdirv[p3 + 0], o_dir + s3 + 0);
  __builtin_nontemporal_store(pdirv[p3 + 1], o_dir + s3 + 1);
  __builtin_nontemporal_store(pdirv[p3 + 2], o_dir + s3 + 2);

  __builtin_nontemporal_store(pconf[p], o_conf + s);

  __builtin_nontemporal_store(xyz[p3 + 0], o_xyz + s3 + 0);
  __builtin_nontemporal_store(xyz[p3 + 1], o_xyz + s3 + 1);
  __builtin_nontemporal_store(xyz[p3 + 2], o_xyz + s3 + 2);

  __builtin_nontemporal_store(pers[p3 + 0], o_pers + s3 + 0);
  __builtin_nontemporal_store(pers[p3 + 1], o_pers + s3 + 1);
  __builtin_nontemporal_store(pers[p3 + 2], o_pers + s3 + 2);

  __builtin_nontemporal_store(mk, o_mask + s);
}

// ---------------------------------------------------------------------------
// Kernel 4: ray_mask int->float, eye(3), vsize zeros, grid_vox_sz scalar.
// ---------------------------------------------------------------------------
__global__ void np_misc_kernel(const int* __restrict__ raymask,
                               float* __restrict__ o_rm, int nrm,
                               float* __restrict__ o_rw2c,
                               float* __restrict__ o_vsize,
                               float* __restrict__ o_grid) {
  const int t = (int)blockIdx.x * blockDim.x + threadIdx.x;
  if (t < nrm)  o_rm[t] = (float)raymask[t];
  if (t < 9)    o_rw2c[t] = ((t % 4) == 0) ? 1.0f : 0.0f;   // eye(3) row-major
  if (t < 3)    o_vsize[t] = 0.0f;
  if (t == 0)   o_grid[0] = 0.0f;
}

extern "C" void kernel_launch(void* const* d_in, const int* in_sizes, int n_in,
                              void* d_out, int out_size, void* d_ws, size_t ws_size,
                              hipStream_t stream) {
  (void)n_in; (void)out_size; (void)ws_size;

  const float* xyz      = (const float*)d_in[0];
  const float* feats    = (const float*)d_in[1];
  const float* pcolor   = (const float*)d_in[2];
  const float* pdirv    = (const float*)d_in[3];
  const float* pconf    = (const float*)d_in[4];
  const float* camrot   = (const float*)d_in[5];
  const float* campos   = (const float*)d_in[6];
  const int*   spidx    = (const int*)d_in[7];
  const float* sloc     = (const float*)d_in[8];
  const float* slocw    = (const float*)d_in[9];
  const float* sraydirs = (const float*)d_in[10];
  const int*   raymask  = (const int*)d_in[11];

  const int npts  = in_sizes[0] / 3;          // 400000
  const int Fdim  = in_sizes[1] / npts;       // 32
  const int slots = in_sizes[7];              // B*R*SR*K = 786432
  const int nrd   = in_sizes[10];             // ray_dirs floats
  const int nrm   = in_sizes[11];             // ray_mask elements

  float* out = (float*)d_out;

  // Output offsets, flat concat in reference return order.
  const size_t o_color = 0;
  const size_t o_rw2c  = o_color + (size_t)slots * 3;
  const size_t o_dir   = o_rw2c + 9;
  const size_t o_conf  = o_dir + (size_t)slots * 3;
  const size_t o_emb   = o_conf + (size_t)slots;
  const size_t o_pers  = o_emb + (size_t)slots * Fdim;
  const size_t o_xyz   = o_pers + (size_t)slots * 3;
  const size_t o_mask  = o_xyz + (size_t)slots * 3;
  const size_t o_loc   = o_mask + (size_t)slots;
  const size_t o_locw  = o_loc + (size_t)slots * 3;
  const size_t o_rd    = o_locw + (size_t)slots * 3;
  const size_t o_rm    = o_rd + (size_t)nrd;
  const size_t o_vs    = o_rm + (size_t)nrm;
  const size_t o_gv    = o_vs + 3;

  float* pers = (float*)d_ws;                 // [N,3] scratch, 4.8 MB

  // 1) w2pers via WMMA: 16 points per wave, 8 waves per 256-thread block.
  {
    const int waves  = (npts + 15) / 16;
    const int blocks = (waves + 7) / 8;
    np_pers_wmma_kernel<<<blocks, 256, 0, stream>>>(xyz, camrot, campos, pers, npts);
  }

  // 2) Embedding gather (F == 32 layout: 8 lanes per slot, float4 chunks).
  {
    const int total  = slots * 8;
    const int blocks = (total + 255) / 256;
    np_gather_emb_kernel<<<blocks, 256, 0, stream>>>(feats, spidx, out + o_emb, slots);
  }

  // 3) Small per-slot gathers (reads pers written by kernel 1; same stream -> ordered).
  {
    const int blocks = (slots + 255) / 256;
    np_gather_misc_kernel<<<blocks, 256, 0, stream>>>(
        spidx, pcolor, pdirv, pconf, xyz, pers,
        out + o_color, out + o_dir, out + o_conf,
        out + o_xyz, out + o_pers, out + o_mask, slots);
  }

  // 4) Constants + ray_mask cast.
  {
    const int threads = nrm > 16 ? nrm : 16;
    const int blocks  = (threads + 255) / 256;
    np_misc_kernel<<<blocks, 256, 0, stream>>>(raymask, out + o_rm, nrm,
                                               out + o_rw2c, out + o_vs, out + o_gv);
  }

  // 5) Pass-through copies (graph-capture-safe async d2d).
  hipMemcpyAsync(out + o_loc,  sloc,     (size_t)slots * 3 * sizeof(float),
                 hipMemcpyDeviceToDevice, stream);
  hipMemcpyAsync(out + o_locw, slocw,    (size_t)slots * 3 * sizeof(float),
                 hipMemcpyDeviceToDevice, stream);
  hipMemcpyAsync(out + o_rd,   sraydirs, (size_t)nrd * sizeof(float),
                 hipMemcpyDeviceToDevice, stream);
}